// GIN_84670985273390
// MI455X (gfx1250) — compile-verified
//
#include <hip/hip_runtime.h>
#include <hip/hip_bf16.h>
#include <stdint.h>

typedef __attribute__((ext_vector_type(2))) float v2f;
typedef __attribute__((ext_vector_type(8))) float v8f;
typedef __attribute__((ext_vector_type(4))) unsigned int v4u;
typedef __attribute__((ext_vector_type(8))) int v8i;
typedef __attribute__((ext_vector_type(4))) int v4i;

#define BN_EPS 1e-5f

#if defined(__has_builtin)
#if __has_builtin(__builtin_amdgcn_tensor_load_to_lds)
#define HAVE_TDM 1
#endif
#endif

// ---------------------------------------------------------------------------
// C[M,N] = A[M,K] @ W[K,N] + bias   (exact f32 via V_WMMA_F32_16X16X4_F32)
// Block: 128 threads = 4 wave32. Block tile: 16 rows x 64 cols.
// A-tile (16 x K) staged into LDS by the Tensor Data Mover (one
// tensor_load_to_lds per block, TENSORcnt-tracked, HW row padding of 1 dword
// per K dwords => LDS row stride K+1 dwords, killing bank conflicts).
// Requires 16 | M (M = 50000 = 16 * 3125).
// ---------------------------------------------------------------------------
template <int K, int N>
__global__ __launch_bounds__(128) void gemm_bias_wmma(
    const float* __restrict__ A, const float* __restrict__ W,
    const float* __restrict__ bias, float* __restrict__ C, int M)
{
    __shared__ float As[16][K + 1];

    const int tid    = threadIdx.x;
    const int wave   = tid >> 5;
    const int lane   = tid & 31;
    const int laneLo = lane & 15;
    const int laneHi = lane >> 4;

    const int rowBase = blockIdx.y * 16;
    const int col     = blockIdx.x * 64 + wave * 16 + laneLo;

#if defined(HAVE_TDM)
    // ---- TDM: DMA the 16 x K f32 tile Global -> LDS with row padding ----
    if (tid == 0) {
        const uint64_t ga   = (uint64_t)(uintptr_t)(A + (size_t)rowBase * K);
        const uint32_t ldsa = (uint32_t)(uintptr_t)(&As[0][0]);  // LDS offset

        v4u g0;
        g0[0] = 1u;                                   // count=1 (valid user D#)
        g0[1] = ldsa;                                 // lds_addr
        g0[2] = (uint32_t)ga;                         // global_addr[31:0]
        g0[3] = (uint32_t)((ga >> 32) & 0x01FFFFFFu)  // global_addr[56:32]
              | (2u << 30);                           // type=2 ("image")

        // data_size=4B (code 2); pad_enable; pad_interval: 128dw->6, 256dw->7;
        // pad_amount code 0 => 1 dword pad per row.
        v8i g1;
        g1[0] = (int)((2u << 16) | (1u << 20) | ((K == 128 ? 6u : 7u) << 22));
        g1[1] = (int)(((uint32_t)K & 0xFFFFu) << 16);            // dim0[15:0] @ [63:48]
        g1[2] = (int)((((uint32_t)K >> 16) & 0xFFFFu)            // dim0[31:16]
              | (((uint32_t)M & 0xFFFFu) << 16));                // dim1[15:0]
        g1[3] = (int)((((uint32_t)M >> 16) & 0xFFFFu)            // dim1[31:16]
              | ((uint32_t)K << 16));                            // tile_dim0=K
        g1[4] = 16;                                              // tile_dim1=16, tile_dim2=0
        g1[5] = K;                                               // dim0_stride[31:0]
        g1[6] = 0;                                               // dim0_stride hi / dim1_stride lo
        g1[7] = 0;
        v4i gz = {0, 0, 0, 0};

#if __clang_major__ >= 23
        v8i gz8 = {0, 0, 0, 0, 0, 0, 0, 0};
        __builtin_amdgcn_tensor_load_to_lds(g0, g1, gz, gz, gz8, 0);
#else
        __builtin_amdgcn_tensor_load_to_lds(g0, g1, gz, gz, 0);
#endif
        __builtin_amdgcn_s_wait_tensorcnt(0);
    }
    __syncthreads();
#else
    // Fallback: cooperative load of the 16 x K tile.
    for (int idx = tid; idx < 16 * K; idx += 128) {
        int r = idx / K;
        int c = idx % K;
        As[r][c] = A[(long long)(rowBase + r) * K + c];
    }
    __syncthreads();
#endif

    // D = A*B + C with C pre-seeded with the bias (bias depends on col only,
    // and every C/D VGPR of a lane maps to the same column).
    float bv = bias[col];
    v8f acc;
#pragma unroll
    for (int i = 0; i < 8; ++i) acc[i] = bv;

    // 32-bit A 16x4 layout: lanes 0-15 -> M=0..15, VGPR0/1 = K={0,1} (lane-lo
    // half) or K={2,3} (lane-hi half). B 4x16 mirrors it across N.
#pragma unroll 4
    for (int k = 0; k < K; k += 4) {
        const int ka = k + 2 * laneHi;
        v2f a, b;
        a.x = As[laneLo][ka];
        a.y = As[laneLo][ka + 1];
        b.x = W[(long long)ka * N + col];
        b.y = W[(long long)(ka + 1) * N + col];
        acc = __builtin_amdgcn_wmma_f32_16x16x4_f32(
            /*neg_a=*/false, a, /*neg_b=*/false, b,
            /*c_mod=*/(short)0, acc, /*reuse_a=*/false, /*reuse_b=*/false);
    }

    // C/D layout: VGPR i -> row i (lanes 0-15) / row i+8 (lanes 16-31).
#pragma unroll
    for (int i = 0; i < 8; ++i) {
        int row = rowBase + 8 * laneHi + i;
        C[(long long)row * N + col] = acc[i];
    }
}

// ---------------------------------------------------------------------------
// agg[dst[e]][c] += x[src[e]][c]  (hardware global_atomic_add_f32)
// ---------------------------------------------------------------------------
__global__ void scatter_add_kernel(const float* __restrict__ x,
                                   const long long* __restrict__ src,
                                   const long long* __restrict__ dst,
                                   float* __restrict__ agg,
                                   int C, long long total)
{
    long long i = (long long)blockIdx.x * blockDim.x + threadIdx.x;
    if (i >= total) return;
    long long e = i / C;
    int c = (int)(i - e * C);
    long long s = src[e];
    long long d = dst[e];
    float v = x[s * C + c];
    unsafeAtomicAdd(&agg[d * C + c], v);
}

// ---------------------------------------------------------------------------
// Deterministic per-column sum / sum-of-squares (one block per column).
// ---------------------------------------------------------------------------
__global__ __launch_bounds__(256) void bn_stats_kernel(
    const float* __restrict__ h, float* __restrict__ sums,
    float* __restrict__ sumsq, int M, int C)
{
    const int col = blockIdx.x;
    const int t   = threadIdx.x;
    float s = 0.f, s2 = 0.f;
    for (int r = t; r < M; r += 256) {
        float v = h[(long long)r * C + col];
        s += v;
        s2 += v * v;
    }
    __shared__ float sh[256], sh2[256];
    sh[t] = s; sh2[t] = s2;
    __syncthreads();
    for (int o = 128; o > 0; o >>= 1) {
        if (t < o) { sh[t] += sh[t + o]; sh2[t] += sh2[t + o]; }
        __syncthreads();
    }
    if (t == 0) { sums[col] = sh[0]; sumsq[col] = sh2[0]; }
}

// h = relu((h - mean) * rsqrt(var + eps) * gamma + beta), in place
__global__ void bn_relu_kernel(float* __restrict__ h,
                               const float* __restrict__ sums,
                               const float* __restrict__ sumsq,
                               const float* __restrict__ gamma,
                               const float* __restrict__ beta,
                               float invM, int C, long long total)
{
    long long i = (long long)blockIdx.x * blockDim.x + threadIdx.x;
    if (i >= total) return;
    int col = (int)(i % C);
    float m   = sums[col] * invM;
    float var = fmaf(-m, m, sumsq[col] * invM);
    float v   = (h[i] - m) * rsqrtf(var + BN_EPS) * gamma[col] + beta[col];
    h[i] = fmaxf(v, 0.f);
}

__global__ void copy4_kernel(const float4* __restrict__ s, float4* __restrict__ d,
                             long long n4)
{
    long long i = (long long)blockIdx.x * blockDim.x + threadIdx.x;
    if (i < n4) d[i] = s[i];
}

// Row-wise log_softmax over 64 columns, one wave32 per row (2 cols/lane).
__global__ __launch_bounds__(128) void log_softmax64_kernel(float* __restrict__ out, int M)
{
    int row  = blockIdx.x * 4 + (threadIdx.x >> 5);
    int lane = threadIdx.x & 31;
    if (row >= M) return;
    float* p = out + (long long)row * 64;
    float a = p[lane], b = p[lane + 32];
    float mx = fmaxf(a, b);
#pragma unroll
    for (int o = 16; o > 0; o >>= 1) mx = fmaxf(mx, __shfl_xor(mx, o, 32));
    float se = expf(a - mx) + expf(b - mx);
#pragma unroll
    for (int o = 16; o > 0; o >>= 1) se += __shfl_xor(se, o, 32);
    float lse = mx + logf(se);
    p[lane]      = a - lse;
    p[lane + 32] = b - lse;
}

// ---------------------------------------------------------------------------
extern "C" void kernel_launch(void* const* d_in, const int* in_sizes, int n_in,
                              void* d_out, int out_size, void* d_ws, size_t ws_size,
                              hipStream_t stream)
{
    const float*     x    = (const float*)d_in[0];
    const long long* ei   = (const long long*)d_in[1];   // int64 in reference
    const float*     W1a  = (const float*)d_in[2];
    const float*     b1a  = (const float*)d_in[3];
    const float*     g1a  = (const float*)d_in[4];
    const float*     be1a = (const float*)d_in[5];
    const float*     W2a  = (const float*)d_in[6];
    const float*     b2a  = (const float*)d_in[7];
    const float*     gout = (const float*)d_in[8];
    const float*     beout= (const float*)d_in[9];
    const float*     W1b  = (const float*)d_in[10];
    const float*     b1b  = (const float*)d_in[11];
    const float*     g1b  = (const float*)d_in[12];
    const float*     be1b = (const float*)d_in[13];
    const float*     W2b  = (const float*)d_in[14];
    const float*     b2b  = (const float*)d_in[15];

    const int       M = in_sizes[0] / 128;     // 50000
    const long long E = in_sizes[1] / 2;       // 800000
    const long long* src = ei;
    const long long* dst = ei + E;

    // Workspace layout: 3 x [M,256] f32 buffers + BN stats (2 x 256 floats).
    float* bufA  = (float*)d_ws;
    float* bufB  = bufA + (size_t)M * 256;
    float* bufC  = bufB + (size_t)M * 256;
    float* sums  = bufC + (size_t)M * 256;
    float* sumsq = sums + 256;

    const float invM = 1.0f / (float)M;
    const int Mt = M / 16;                    // 3125 row tiles

    auto g1d = [](long long n) { return (unsigned)((n + 255) / 256); };

    // ---- Layer 1: agg1 = x + segment_sum(x[src] -> dst), in bufC [M,128]
    copy4_kernel<<<g1d((long long)M * 128 / 4), 256, 0, stream>>>(
        (const float4*)x, (float4*)bufC, (long long)M * 128 / 4);
    scatter_add_kernel<<<g1d(E * 128), 256, 0, stream>>>(x, src, dst, bufC, 128, E * 128);

    // h1 = agg1 @ W1a + b1a   -> bufA [M,256]
    gemm_bias_wmma<128, 256><<<dim3(4, Mt), 128, 0, stream>>>(bufC, W1a, b1a, bufA, M);
    bn_stats_kernel<<<256, 256, 0, stream>>>(bufA, sums, sumsq, M, 256);
    bn_relu_kernel<<<g1d((long long)M * 256), 256, 0, stream>>>(
        bufA, sums, sumsq, g1a, be1a, invM, 256, (long long)M * 256);

    // h2 = h1 @ W2a + b2a     -> bufB [M,256] ; outer BN + ReLU
    gemm_bias_wmma<256, 256><<<dim3(4, Mt), 128, 0, stream>>>(bufA, W2a, b2a, bufB, M);
    bn_stats_kernel<<<256, 256, 0, stream>>>(bufB, sums, sumsq, M, 256);
    bn_relu_kernel<<<g1d((long long)M * 256), 256, 0, stream>>>(
        bufB, sums, sumsq, gout, beout, invM, 256, (long long)M * 256);

    // ---- Layer 2: agg2 = h2 + segment_sum(h2[src] -> dst), in bufA [M,256]
    copy4_kernel<<<g1d((long long)M * 256 / 4), 256, 0, stream>>>(
        (const float4*)bufB, (float4*)bufA, (long long)M * 256 / 4);
    scatter_add_kernel<<<g1d(E * 256), 256, 0, stream>>>(bufB, src, dst, bufA, 256, E * 256);

    // h3 = agg2 @ W1b + b1b   -> bufC [M,256]
    gemm_bias_wmma<256, 256><<<dim3(4, Mt), 128, 0, stream>>>(bufA, W1b, b1b, bufC, M);
    bn_stats_kernel<<<256, 256, 0, stream>>>(bufC, sums, sumsq, M, 256);
    bn_relu_kernel<<<g1d((long long)M * 256), 256, 0, stream>>>(
        bufC, sums, sumsq, g1b, be1b, invM, 256, (long long)M * 256);

    // out = h3 @ W2b + b2b    -> d_out [M,64] ; then log_softmax in place
    gemm_bias_wmma<256, 64><<<dim3(1, Mt), 128, 0, stream>>>(bufC, W2b, b2b, (float*)d_out, M);
    log_softmax64_kernel<<<(M + 3) / 4, 128, 0, stream>>>((float*)d_out, M);
}